// U1_circuit_73942156968354
// MI455X (gfx1250) — compile-verified
//
#include <hip/hip_runtime.h>

typedef float v2f __attribute__((ext_vector_type(2)));
typedef float v8f __attribute__((ext_vector_type(8)));

#define PI_F        3.14159265358979323846f
#define INV_SQRT2_F 0.70710678118654752f

// ---- 5-qubit state helpers: 32 complex amplitudes in registers ----
// Qubit q <-> bit (1 << (4 - q)) of the flat amplitude index.

// RX(theta): [[c, -i s], [-i s, c]] on qubit with mask M (c=cos(th/2), s=sin(th/2))
template <int M>
__device__ __forceinline__ void rx_gate(float* sr, float* si, float c, float s) {
#pragma unroll
  for (int k = 0; k < 32; ++k) {
    if ((k & M) == 0) {
      const int k2 = k | M;
      float ar = sr[k], ai = si[k];
      float br = sr[k2], bi = si[k2];
      sr[k]  =  c * ar + s * bi;
      si[k]  =  c * ai - s * br;
      sr[k2] =  s * ai + c * br;
      si[k2] = -s * ar + c * bi;
    }
  }
}

// CX^p on (control mask MC, target mask MT): on the control=1 subspace apply
// [[a, b], [b, a]] with a=(1+lam)/2, b=(1-lam)/2, lam=exp(i*pi*p).
template <int MC, int MT>
__device__ __forceinline__ void cxpow_gate(float* sr, float* si,
                                           float ar_, float ai_, float br_, float bi_) {
#pragma unroll
  for (int k = 0; k < 32; ++k) {
    if (((k & MC) != 0) && ((k & MT) == 0)) {
      const int k2 = k | MT;
      float xr = sr[k],  xi = si[k];
      float yr = sr[k2], yi = si[k2];
      sr[k]  = ar_ * xr - ai_ * xi + br_ * yr - bi_ * yi;
      si[k]  = ar_ * xi + ai_ * xr + br_ * yi + bi_ * yr;
      sr[k2] = br_ * xr - bi_ * xi + ar_ * yr - ai_ * yi;
      si[k2] = br_ * xi + bi_ * xr + ar_ * yi + ai_ * yr;
    }
  }
}

// CZ^p on qubits (1,0): multiply amplitudes with bits 8 and 16 set by lam.
__device__ __forceinline__ void czpow_gate(float* sr, float* si, float lr, float li) {
#pragma unroll
  for (int k = 0; k < 32; ++k) {
    if ((k & 24) == 24) {
      float xr = sr[k], xi = si[k];
      sr[k] = xr * lr - xi * li;
      si[k] = xr * li + xi * lr;
    }
  }
}

// One thread per (pixel, filter). 246016 = 961 blocks * 256 threads (exact).
__global__ __launch_bounds__(256) void qconv31_kernel(const float* __restrict__ in,
                                                      const float* __restrict__ ker,
                                                      float* __restrict__ out,
                                                      int total) {
  const int tid = blockIdx.x * 256 + threadIdx.x;
  if (tid >= total) return;  // never taken (exact grid), EXEC stays all-ones

  const int f  = tid & 7;        // filter (output is innermost over filters)
  const int r  = tid >> 3;       // flat (b, iy, ix)
  const int ix = r % 31;
  const int ry = r / 31;
  const int iy = ry % 31;
  const int b  = ry / 31;

  // RX coefficients: angle index = 4*channel + corner; theta = pi*angle
  float cA[12], sA[12];
#pragma unroll
  for (int g = 0; g < 4; ++g) {
    const int dh = g >> 1, dw = g & 1;
    const int base = (((b * 32) + iy + dh) * 32 + (ix + dw)) * 3;
#pragma unroll
    for (int c = 0; c < 3; ++c) {
      float s, cc;
      __sincosf(0.5f * PI_F * in[base + c], &s, &cc);
      cA[4 * c + g] = cc;
      sA[4 * c + g] = s;
    }
  }

  const float* prm = ker + f * 15;  // kernel[f, 0, :]

  // |00000> then H on qubit 0
  float sr[32], si[32];
#pragma unroll
  for (int k = 0; k < 32; ++k) { sr[k] = 0.0f; si[k] = 0.0f; }
  sr[0]  = INV_SQRT2_F;
  sr[16] = INV_SQRT2_F;

#pragma unroll
  for (int j = 0; j < 3; ++j) {
    rx_gate<8>(sr, si, cA[4 * j + 0], sA[4 * j + 0]);  // qubit 1
    rx_gate<4>(sr, si, cA[4 * j + 1], sA[4 * j + 1]);  // qubit 2
    rx_gate<2>(sr, si, cA[4 * j + 2], sA[4 * j + 2]);  // qubit 3
    rx_gate<1>(sr, si, cA[4 * j + 3], sA[4 * j + 3]);  // qubit 4

    float lr[5], li[5];
#pragma unroll
    for (int g = 0; g < 5; ++g)
      __sincosf(PI_F * prm[5 * j + g], &li[g], &lr[g]);  // lam = exp(i*pi*p)

#define CXP(G, MC, MT)                                                        \
    cxpow_gate<MC, MT>(sr, si, 0.5f * (1.0f + lr[G]),  0.5f * li[G],          \
                               0.5f * (1.0f - lr[G]), -0.5f * li[G])
    CXP(0, 8, 4);  // (1,2)
    CXP(1, 4, 2);  // (2,3)
    CXP(2, 2, 1);  // (3,4)
    CXP(3, 1, 8);  // (4,1)
#undef CXP
    czpow_gate(sr, si, lr[4], li[4]);  // CZ^p on (1,0)
  }

  // ---- <X0> = 2 * sum_k Re(conj(lo_k) * hi_k): length-32 real dot product.
  // Batch the wave's 32 dot products through v_wmma_f32_16x16x4_f32:
  // diag(A * B) per 16 sims, accumulated over 8 K-chunks of 4.
  float vv[32], ww[32];
#pragma unroll
  for (int k = 0; k < 16; ++k) {
    vv[k]      = sr[k];       // lo real
    vv[k + 16] = si[k];       // lo imag
    ww[k]      = sr[k + 16];  // hi real
    ww[k + 16] = si[k + 16];  // hi imag
  }

  const int  lane = threadIdx.x & 31;
  const bool lo   = lane < 16;

  v8f acc0 = {};  // sims = lanes 0..15
  v8f acc1 = {};  // sims = lanes 16..31
#pragma unroll
  for (int c = 0; c < 8; ++c) {
    const int k0 = 4 * c;
    // Group 0: A[m][k]=vv_m[k0+k], B[k][n]=ww_n[k0+k], m,n = lanes 0..15.
    // A/B VGPR0 holds K={0 | 2} (lower|upper half-wave), VGPR1 holds K={1 | 3}.
    float a2 = __shfl(vv[k0 + 2], lane & 15, 32);
    float a3 = __shfl(vv[k0 + 3], lane & 15, 32);
    float b2 = __shfl(ww[k0 + 2], lane & 15, 32);
    float b3 = __shfl(ww[k0 + 3], lane & 15, 32);
    v2f A0 = { lo ? vv[k0 + 0] : a2, lo ? vv[k0 + 1] : a3 };
    v2f B0 = { lo ? ww[k0 + 0] : b2, lo ? ww[k0 + 1] : b3 };
    acc0 = __builtin_amdgcn_wmma_f32_16x16x4_f32(false, A0, false, B0,
                                                 (short)0, acc0, false, false);
    // Group 1: sims = lanes 16..31 (lower half gathers from lane+16).
    float g0 = __shfl(vv[k0 + 0], (lane & 15) + 16, 32);
    float g1 = __shfl(vv[k0 + 1], (lane & 15) + 16, 32);
    float h0 = __shfl(ww[k0 + 0], (lane & 15) + 16, 32);
    float h1 = __shfl(ww[k0 + 1], (lane & 15) + 16, 32);
    v2f A1 = { lo ? g0 : vv[k0 + 2], lo ? g1 : vv[k0 + 3] };
    v2f B1 = { lo ? h0 : ww[k0 + 2], lo ? h1 : ww[k0 + 3] };
    acc1 = __builtin_amdgcn_wmma_f32_16x16x4_f32(false, A1, false, B1,
                                                 (short)0, acc1, false, false);
  }

  // Diagonal extraction. D layout: lanes 0-15 VGPR v = D[v][lane];
  // lanes 16-31 VGPR v = D[v+8][lane-16]. So D[t][t] lives at:
  //   t in 0..7  -> lane t,      VGPR t
  //   t in 8..15 -> lane t+16,   VGPR t-8
  const int vi = lane & 7;
  float d0 = acc0[0], d1 = acc1[0];
#pragma unroll
  for (int i = 1; i < 8; ++i) {
    d0 = (vi == i) ? acc0[i] : d0;
    d1 = (vi == i) ? acc1[i] : d1;
  }
  const int s15 = lane & 15;
  const int src = (s15 < 8) ? s15 : (s15 + 16);
  float e0 = __shfl(d0, src, 32);  // dots for sims 0..15 -> lanes 0..15
  float e1 = __shfl(d1, src, 32);  // dots for sims 16..31 -> lanes 0..15 pattern
  float e  = 2.0f * (lo ? e0 : e1);

  float cl = fminf(fmaxf(e, -1.0f + 1e-5f), 1.0f - 1e-5f);
  out[tid] = acosf(cl) * (1.0f / PI_F);
}

extern "C" void kernel_launch(void* const* d_in, const int* in_sizes, int n_in,
                              void* d_out, int out_size, void* d_ws, size_t ws_size,
                              hipStream_t stream) {
  const float* in  = (const float*)d_in[0];   // (32,32,32,3) f32
  const float* ker = (const float*)d_in[1];   // (8,1,15)     f32
  float* out = (float*)d_out;                 // (32,31,31,8) f32 = 246016
  const int total = out_size;                 // 246016 = 961 * 256 exactly
  const int block = 256;
  const int grid  = (total + block - 1) / block;
  qconv31_kernel<<<grid, block, 0, stream>>>(in, ker, out, total);
}